// FastNDCG_14611478740992
// MI455X (gfx1250) — compile-verified
//
#include <hip/hip_runtime.h>
#include <stdint.h>

// ---------------------------------------------------------------------------
// FastNDCG on MI455X (gfx1250).
// Phase B (first in file so the disasm snippet shows it): stream 8M int32
//   indexes through LDS via the Tensor Data Mover (tensor_load_to_lds +
//   s_wait_tensorcnt, double-buffered 16KB tiles); waves spend issue slots on
//   ds_load + global_atomic_min/max only. Falls back to int4 loads if the
//   TDM builtin is unavailable.
// Phase A: init first/last arrays. Phase C: per-user closed-form dcg/idcg +
//   f32 block tree reduce. Phase D: single-block f64 reduce -> mean.
// All integer atomics => fully deterministic across graph replays.
// ---------------------------------------------------------------------------

#define NDCG_USERS 400000
#define NDCG_NEG_INF (-1.0e9f)
#define NDCG_LOG2_3 1.5849625007211562f /* rounds to the f32 XLA uses */
#define NDCG_BLOCK 256
#define NDCG_TILE_INTS 4096 /* 16KB per LDS tile, x2 buffers = 32KB */
#define NDCG_SCATTER_GRID 512

#if defined(__has_include)
#if __has_include(<hip/amd_detail/amd_gfx1250_TDM.h>)
#define NDCG_TDM_ARITY6 1
#else
#define NDCG_TDM_ARITY6 0
#endif
#else
#define NDCG_TDM_ARITY6 0
#endif

#if defined(__HIP_DEVICE_COMPILE__) && defined(__gfx1250__)
#if __has_builtin(__builtin_amdgcn_tensor_load_to_lds) && \
    __has_builtin(__builtin_amdgcn_s_wait_tensorcnt)
#define NDCG_USE_TDM 1
#endif
#endif
#ifndef NDCG_USE_TDM
#define NDCG_USE_TDM 0
#endif

typedef unsigned int ndcg_v4u __attribute__((ext_vector_type(4)));
typedef int ndcg_v4i __attribute__((ext_vector_type(4)));
typedef int ndcg_v8i __attribute__((ext_vector_type(8)));

#if NDCG_USE_TDM
// Issue one 1-D TDM tile load: elems int32 from gsrc -> LDS at ldsdst.
// Descriptor packing per CDNA5 ISA §8.3 (group0) / §8.4 (group1).
__device__ __forceinline__ void ndcg_tdm_issue(const int* gsrc, void* ldsdst,
                                               int elems) {
  unsigned long long ga = (unsigned long long)(size_t)gsrc;
  unsigned lds_off = (unsigned)(size_t)ldsdst; // low 32 bits = LDS byte offset
  ndcg_v4u g0;
  g0.x = 1u;                 // count=1 valid descriptor, no gather mode
  g0.y = lds_off;            // lds_addr
  g0.z = (unsigned)ga;       // global_addr[31:0]
  g0.w = ((unsigned)(ga >> 32) & 0x01FFFFFFu) | 0x80000000u; // addr[56:32]|type=2
  unsigned td0 = (unsigned)elems;
  ndcg_v8i g1;
  g1[0] = (int)(2u << 16);                 // data_size = 4 bytes
  g1[1] = (int)((td0 & 0xFFFFu) << 16);    // tensor_dim0[15:0]
  g1[2] = (int)((td0 >> 16) | (1u << 16)); // tensor_dim0[31:16] | tensor_dim1=1
  g1[3] = (int)((td0 & 0xFFFFu) << 16);    // tile_dim0 = elems (<= 4096)
  g1[4] = 0;                               // tile_dim1 = tile_dim2 = 0 (unused)
  g1[5] = (int)td0;                        // tensor_dim0_stride (irrelevant, 1D)
  g1[6] = 0;
  g1[7] = 0;
  ndcg_v4i gz = (ndcg_v4i)0;
#if NDCG_TDM_ARITY6
  ndcg_v8i gz8 = (ndcg_v8i)0;
  __builtin_amdgcn_tensor_load_to_lds(g0, g1, gz, gz, gz8, 0);
#else
  __builtin_amdgcn_tensor_load_to_lds(g0, g1, gz, gz, 0);
#endif
}
#endif // NDCG_USE_TDM

__global__ __launch_bounds__(NDCG_BLOCK) void ndcg_scatter(
    const int* __restrict__ idx, int n, int* __restrict__ fp,
    int* __restrict__ lp) {
#if NDCG_USE_TDM
  __shared__ int buf[2][NDCG_TILE_INTS];
  const int G = gridDim.x;
  const int numTiles = (n + NDCG_TILE_INTS - 1) / NDCG_TILE_INTS;
  const int b = blockIdx.x;
  const int mine = (b < numTiles) ? ((numTiles - 1 - b) / G + 1) : 0;
  const bool issuer = (threadIdx.x < 32); // wave 0 owns the TDM queue

  if (issuer && mine > 0) {
    long base0 = (long)b * NDCG_TILE_INTS;
    long rem0 = (long)n - base0;
    int e0 = rem0 < (long)NDCG_TILE_INTS ? (int)rem0 : NDCG_TILE_INTS;
    ndcg_tdm_issue(idx + base0, &buf[0][0], e0);
    if (mine > 1) {
      long base1 = (long)(b + G) * NDCG_TILE_INTS;
      long rem1 = (long)n - base1;
      int e1 = rem1 < (long)NDCG_TILE_INTS ? (int)rem1 : NDCG_TILE_INTS;
      ndcg_tdm_issue(idx + base1, &buf[1][0], e1);
    }
  }
  for (int k = 0; k < mine; ++k) {
    long base = (long)(b + (long)k * G) * NDCG_TILE_INTS;
    long rem = (long)n - base;
    int elems = rem < (long)NDCG_TILE_INTS ? (int)rem : NDCG_TILE_INTS;
    if (issuer) {
      if (k + 1 < mine) {
        __builtin_amdgcn_s_wait_tensorcnt(1); // oldest tile landed
      } else {
        __builtin_amdgcn_s_wait_tensorcnt(0); // last tile: drain
      }
    }
    __syncthreads();
    const int* t = &buf[k & 1][0];
    for (int j = (int)threadIdx.x; j < elems; j += NDCG_BLOCK) {
      int u = t[j];
      int pos = (int)(base + j);
      atomicMin(&fp[u], pos);
      atomicMax(&lp[u], pos);
    }
    __syncthreads(); // buffer fully consumed before refill
    if (issuer && (k + 2 < mine)) {
      long b2 = (long)(b + (long)(k + 2) * G) * NDCG_TILE_INTS;
      long r2 = (long)n - b2;
      int e2 = r2 < (long)NDCG_TILE_INTS ? (int)r2 : NDCG_TILE_INTS;
      ndcg_tdm_issue(idx + b2, &buf[k & 1][0], e2);
    }
  }
#else
  // Fallback: coalesced int4 streaming loads + atomics.
  const int4* v = (const int4*)idx;
  int n4 = n >> 2;
  int stride = gridDim.x * blockDim.x;
  for (int i = blockIdx.x * blockDim.x + threadIdx.x; i < n4; i += stride) {
    int4 u = v[i];
    int base = i << 2;
    atomicMin(&fp[u.x], base + 0);
    atomicMax(&lp[u.x], base + 0);
    atomicMin(&fp[u.y], base + 1);
    atomicMax(&lp[u.y], base + 1);
    atomicMin(&fp[u.z], base + 2);
    atomicMax(&lp[u.z], base + 2);
    atomicMin(&fp[u.w], base + 3);
    atomicMax(&lp[u.w], base + 3);
  }
  if (blockIdx.x == 0 && threadIdx.x == 0) {
    for (int i = n4 << 2; i < n; ++i) {
      atomicMin(&fp[idx[i]], i);
      atomicMax(&lp[idx[i]], i);
    }
  }
#endif
}

__global__ void ndcg_init(int* __restrict__ fp, int* __restrict__ lp) {
  int i = blockIdx.x * blockDim.x + threadIdx.x;
  if (i < NDCG_USERS) {
    fp[i] = 0x7FFFFFFF;
    lp[i] = -1;
  }
}

__global__ __launch_bounds__(NDCG_BLOCK) void ndcg_user(
    const float* __restrict__ pred, const float* __restrict__ targ,
    const int* __restrict__ fp, const int* __restrict__ lp,
    float* __restrict__ partials, int n) {
  int u = blockIdx.x * blockDim.x + threadIdx.x;
  float ratio = 0.0f;
  if (u < NDCG_USERS) {
    int f = fp[u];
    int l = lp[u];
    bool has2 = (l > f); // positions unique -> count>=2
    int fi = f;
    if (fi > n - 1) fi = n - 1; // replicate JAX clamped gather for empty users
    float p0 = pred[fi];
    float t0 = targ[fi];
    float p1 = has2 ? pred[l] : NDCG_NEG_INF;
    float t1 = has2 ? targ[l] : 0.0f;
    // top-k is stable: tie p0==p1 ranks index 0 first.
    float dcg = (p0 >= p1) ? (t0 + t1 / NDCG_LOG2_3) : (t1 + t0 / NDCG_LOG2_3);
    float hi = fmaxf(t0, t1);
    float lo = fminf(t0, t1);
    float idcg = hi + lo / NDCG_LOG2_3;
    ratio = dcg / idcg; // 0/0 -> NaN, matching the reference
  }
  __shared__ float s[NDCG_BLOCK];
  s[threadIdx.x] = ratio;
  __syncthreads();
  for (int off = NDCG_BLOCK / 2; off > 0; off >>= 1) {
    if ((int)threadIdx.x < off) s[threadIdx.x] += s[threadIdx.x + off];
    __syncthreads();
  }
  if (threadIdx.x == 0) partials[blockIdx.x] = s[0];
}

__global__ __launch_bounds__(NDCG_BLOCK) void ndcg_final(
    const float* __restrict__ partials, int nparts, float* __restrict__ out) {
  __shared__ double s[NDCG_BLOCK];
  double acc = 0.0;
  for (int i = (int)threadIdx.x; i < nparts; i += NDCG_BLOCK)
    acc += (double)partials[i];
  s[threadIdx.x] = acc;
  __syncthreads();
  for (int off = NDCG_BLOCK / 2; off > 0; off >>= 1) {
    if ((int)threadIdx.x < off) s[threadIdx.x] += s[threadIdx.x + off];
    __syncthreads();
  }
  if (threadIdx.x == 0) out[0] = (float)(s[0] / (double)NDCG_USERS);
}

extern "C" void kernel_launch(void* const* d_in, const int* in_sizes, int n_in,
                              void* d_out, int out_size, void* d_ws,
                              size_t ws_size, hipStream_t stream) {
  (void)n_in;
  (void)out_size;
  (void)ws_size;
  const float* pred = (const float*)d_in[0];
  const float* targ = (const float*)d_in[1];
  const int* idx = (const int*)d_in[2]; // int32 per harness convention
  const int n = in_sizes[0];

  int* fp = (int*)d_ws;                    // NDCG_USERS ints
  int* lp = fp + NDCG_USERS;               // NDCG_USERS ints
  float* partials = (float*)(lp + NDCG_USERS);

  const int gridU = (NDCG_USERS + NDCG_BLOCK - 1) / NDCG_BLOCK; // 1563

  ndcg_init<<<gridU, NDCG_BLOCK, 0, stream>>>(fp, lp);
  ndcg_scatter<<<NDCG_SCATTER_GRID, NDCG_BLOCK, 0, stream>>>(idx, n, fp, lp);
  ndcg_user<<<gridU, NDCG_BLOCK, 0, stream>>>(pred, targ, fp, lp, partials, n);
  ndcg_final<<<1, NDCG_BLOCK, 0, stream>>>(partials, gridU, (float*)d_out);
}